// TotalLoss_2800318677529
// MI455X (gfx1250) — compile-verified
//
#include <hip/hip_runtime.h>
#include <math.h>

// Problem constants (from reference)
#define B_SZ 4096
#define S_SZ 200
#define V_SZ 33     // data feature dim; long_pred has V_SZ-1 = 32
#define T_SZ 50
#define EPSV 1e-8f
#define NWAVES 8    // 256 threads / wave32

typedef __attribute__((ext_vector_type(2))) float v2f;
typedef __attribute__((ext_vector_type(8))) float v8f;

// ---------------------------------------------------------------------------
// Full wave32 sum using V_WMMA_F32_16X16X4_F32.
//   A (16x4): lane L<16 holds A[L,0]=p, A[L,1]=0; lane L>=16 holds A[L-16,2]=p.
//   B (4x16): all ones  =>  D[m,n] = p[m] + p[m+16] for every n.
//   Lane-local sum of the 8 D regs gives sum over m=0..7 (lanes 0-15) /
//   m=8..15 (lanes 16-31); one xor-16 shuffle completes the 32-lane sum.
// Requires EXEC == all ones (call only from fully convergent code).
// ---------------------------------------------------------------------------
__device__ __forceinline__ float wave_sum32(float p) {
  v2f a;  a[0] = p;    a[1] = 0.0f;
  v2f bo; bo[0] = 1.0f; bo[1] = 1.0f;
  v8f c = {};
  c = __builtin_amdgcn_wmma_f32_16x16x4_f32(
      /*neg_a=*/false, a, /*neg_b=*/false, bo,
      /*c_mod=*/(short)0, c, /*reuse_a=*/false, /*reuse_b=*/false);
  float t = ((c[0] + c[1]) + (c[2] + c[3])) + ((c[4] + c[5]) + (c[6] + c[7]));
  t += __shfl_xor(t, 16, 32);
  return t;
}

// ---------------------------------------------------------------------------
// Workspace init (harness poisons d_ws once and never re-poisons).
// layout: g[0] = sum of squared diffs (double), g[1] = sum log f (double),
//         *(int*)(ws+16) = total mask count
// ---------------------------------------------------------------------------
__global__ void init_ws(double* __restrict__ g, int* __restrict__ mcnt) {
  if (threadIdx.x == 0) { g[0] = 0.0; g[1] = 0.0; *mcnt = 0; }
}

// ---------------------------------------------------------------------------
// Longitudinal loss: one block (8 waves) per row b.
// Phase 1: ballot-based valid count + deterministic stream compaction of the
//          masked s-indices into LDS (no atomics, ordered).
// Phase 2: branch-free loop over compacted indices only (~50% of rows on
//          average -> halves HBM traffic vs dense), manually unrolled x4 so
//          each wave keeps 8 independent 128B-coalesced loads in flight.
//          Scalar (SGPR) base + 32-bit lane offsets -> GVS addressing, no
//          per-lane 64-bit address arithmetic in the hot loop.
// ---------------------------------------------------------------------------
__global__ __launch_bounds__(256) void long_loss_kernel(
    const float* __restrict__ long_pred, const float* __restrict__ data,
    double* __restrict__ g_sumsq, int* __restrict__ g_mcnt) {
  __shared__ unsigned s_vbal[NWAVES];   // valid-flag ballots
  __shared__ unsigned s_mbal[NWAVES];   // hist-mask ballots
  __shared__ short    s_list[S_SZ];     // compacted masked s indices
  __shared__ float    s_wsum[NWAVES];

  const int b    = blockIdx.x;
  const int tid  = threadIdx.x;
  const int wave = tid >> 5;
  const int lane = tid & 31;

  // Phase 1a: valid[s] = !isnan(data[b,s,1]); len = popcount over all s
  bool v = false;
  if (tid < S_SZ) {
    const float x0 = data[(unsigned)(b * S_SZ + tid) * V_SZ + 1u];
    v = !__builtin_isnan(x0);
  }
  const unsigned vb = (unsigned)__ballot(v);
  if (lane == 0) s_vbal[wave] = vb;
  __syncthreads();

  int len = 0;
#pragma unroll
  for (int w = 0; w < NWAVES; ++w) len += __popc(s_vbal[w]);
  const int idx_last = len - 1;

  // Phase 1b: m[s] = valid[s] && s != idx_last, over s in [0, S-1);
  // compact indices with an ordered ballot prefix-sum.
  const bool m = v && (tid < S_SZ - 1) && (tid != idx_last);
  const unsigned mb = (unsigned)__ballot(m);
  if (lane == 0) s_mbal[wave] = mb;
  __syncthreads();

  int nm = 0, base = 0;
#pragma unroll
  for (int w = 0; w < NWAVES; ++w) {
    const int c = __popc(s_mbal[w]);
    if (w < wave) base += c;
    nm += c;
  }
  if (m) {
    const int pos = base + __popc(mb & ((1u << lane) - 1));
    s_list[pos] = (short)tid;
  }
  __syncthreads();

  // Phase 2: acc += (long_pred[b,s,v] - data[b,s+1,v+1])^2 over compacted s.
  // lane v <-> feature v: each load is 32 lanes x 4B = 128B contiguous.
  // 32-bit element offsets (arrays are ~27M elements < 2^31).
  const unsigned lp_base = (unsigned)(b * S_SZ) * (V_SZ - 1) + (unsigned)lane;
  const unsigned d_base  = (unsigned)(b * S_SZ + 1) * V_SZ + 1u + (unsigned)lane;

  float acc = 0.0f;
  int k = wave;
  for (; k + 3 * NWAVES < nm; k += 4 * NWAVES) {
    const unsigned s0 = (unsigned)s_list[k];
    const unsigned s1 = (unsigned)s_list[k + NWAVES];
    const unsigned s2 = (unsigned)s_list[k + 2 * NWAVES];
    const unsigned s3 = (unsigned)s_list[k + 3 * NWAVES];
    const float a0 = long_pred[lp_base + s0 * (V_SZ - 1)];
    const float c0 = data[d_base + s0 * V_SZ];
    const float a1 = long_pred[lp_base + s1 * (V_SZ - 1)];
    const float c1 = data[d_base + s1 * V_SZ];
    const float a2 = long_pred[lp_base + s2 * (V_SZ - 1)];
    const float c2 = data[d_base + s2 * V_SZ];
    const float a3 = long_pred[lp_base + s3 * (V_SZ - 1)];
    const float c3 = data[d_base + s3 * V_SZ];
    const float d0 = a0 - c0, d1 = a1 - c1, d2 = a2 - c2, d3 = a3 - c3;
    acc = fmaf(d0, d0, acc);
    acc = fmaf(d1, d1, acc);
    acc = fmaf(d2, d2, acc);
    acc = fmaf(d3, d3, acc);
  }
  for (; k < nm; k += NWAVES) {
    const unsigned s = (unsigned)s_list[k];
    const float d = long_pred[lp_base + s * (V_SZ - 1)] - data[d_base + s * V_SZ];
    acc = fmaf(d, d, acc);
  }

  // wave32 reduction on the matrix pipe (fully convergent here)
  const float wsum = wave_sum32(acc);
  if (lane == 0) s_wsum[wave] = wsum;
  __syncthreads();

  if (tid == 0) {
    float tot = 0.0f;
#pragma unroll
    for (int w = 0; w < NWAVES; ++w) tot += s_wsum[w];
    atomicAdd(g_sumsq, (double)tot);
    atomicAdd(g_mcnt, nm);
  }
}

// ---------------------------------------------------------------------------
// Survival NLL: one thread per b; grid covers exactly B, so the WMMA wave
// reduction sees EXEC = all ones.
// ---------------------------------------------------------------------------
__global__ __launch_bounds__(256) void nll_kernel(
    const float* __restrict__ surv, const int* __restrict__ event,
    const float* __restrict__ etime, const float* __restrict__ trange,
    double* __restrict__ g_logsum) {
  __shared__ float s_wsum[NWAVES];
  const int tid = threadIdx.x;
  const int b   = blockIdx.x * blockDim.x + tid;

  const float et = etime[b];
  // searchsorted(trange, et, 'right') - 1 == count(trange <= et) - 1
  int idx = -1;
#pragma unroll
  for (int t = 0; t < T_SZ; ++t) idx += (trange[t] <= et) ? 1 : 0;
  if (idx < 0) idx = 0;

  const unsigned rbase = (unsigned)b * T_SZ;
  float f;
  if (event[b] != 0) {
    f = surv[rbase + (unsigned)idx];
  } else {
    float tail = 0.0f;  // rc[idx+1] = sum_{t>=idx+1} surv[t]; 0 if idx+1 >= T
    for (int t = idx + 1; t < T_SZ; ++t) tail += surv[rbase + (unsigned)t];
    f = tail;
  }
  if (f == 0.0f) f = EPSV;
  const float lg = logf(f);

  const float wsum = wave_sum32(lg);
  if ((tid & 31) == 0) s_wsum[tid >> 5] = wsum;
  __syncthreads();
  if (tid == 0) {
    float tot = 0.0f;
#pragma unroll
    for (int w = 0; w < NWAVES; ++w) tot += s_wsum[w];
    atomicAdd(g_logsum, (double)tot);
  }
}

// ---------------------------------------------------------------------------
// Combine: out = -mean(log f) + sumsq / (mcount * 32)
// denom as float(count)*32 matches the reference f32 cast; exact since
// count < 2^24 and x32 is a pure exponent shift.
// ---------------------------------------------------------------------------
__global__ void combine_kernel(const double* __restrict__ g,
                               const int* __restrict__ mcnt,
                               float* __restrict__ out) {
  if (threadIdx.x == 0 && blockIdx.x == 0) {
    const double nll   = -(g[1] / (double)B_SZ);
    const float  denom = (float)(*mcnt) * (float)(V_SZ - 1);
    const double ll    = g[0] / (double)denom;
    out[0] = (float)(nll + ll);
  }
}

extern "C" void kernel_launch(void* const* d_in, const int* in_sizes, int n_in,
                              void* d_out, int out_size, void* d_ws, size_t ws_size,
                              hipStream_t stream) {
  // setup_inputs() order: long_pred, surv_pred, data, event, event_time, time_range
  const float* long_pred = (const float*)d_in[0];
  const float* surv      = (const float*)d_in[1];
  const float* data      = (const float*)d_in[2];
  const int*   event     = (const int*)d_in[3];   // bool -> integer per harness
  const float* etime     = (const float*)d_in[4];
  const float* trange    = (const float*)d_in[5];

  double* g    = (double*)d_ws;                 // g[0]=sumsq, g[1]=logsum
  int*    mcnt = (int*)((char*)d_ws + 16);
  float*  out  = (float*)d_out;

  hipLaunchKernelGGL(init_ws, dim3(1), dim3(32), 0, stream, g, mcnt);
  hipLaunchKernelGGL(long_loss_kernel, dim3(B_SZ), dim3(256), 0, stream,
                     long_pred, data, g, mcnt);
  hipLaunchKernelGGL(nll_kernel, dim3(B_SZ / 256), dim3(256), 0, stream,
                     surv, event, etime, trange, g + 1);
  hipLaunchKernelGGL(combine_kernel, dim3(1), dim3(32), 0, stream, g, mcnt, out);
}